// Propagation2_64974265254027
// MI455X (gfx1250) — compile-verified
//
#include <hip/hip_runtime.h>
#include <cstdint>

// Propagation2: out[b,k,y,x] = in[b,0, clamp(y+dy[k]), clamp(x+dx[k])]
// (9 one-hot 5x5 filters over ReplicationPad2d(2) == 9 clamped shifts)
//
// Pure data movement: 16.7 MB in (L2-resident), 150.4 MB out -> HBM-store bound
// (~7.2 us floor at 23.3 TB/s). CDNA5 path: async global->LDS DMA (ASYNCcnt)
// stages tile+halo with clamped addresses, then 9 shifted LDS reads ->
// aligned non-temporal b128 stores.

typedef float v4f __attribute__((ext_vector_type(4)));

#define TILE_W 64
#define TILE_H 16
#define HALO   2
#define LW     (TILE_W + 2 * HALO)   // 68 floats per LDS row
#define LH     (TILE_H + 2 * HALO)   // 20 LDS rows
#define LDS_ELEMS (LW * LH)          // 1360 floats = 5440 B
#define NTHREADS 256
#define NITER   ((LDS_ELEMS + NTHREADS - 1) / NTHREADS)   // 6

__device__ __forceinline__ void async_f32_to_lds(const float* gaddr, uint32_t lds_byte_addr) {
#if __has_builtin(__builtin_amdgcn_global_load_async_to_lds_b32)
    typedef __attribute__((address_space(1))) int gas_int;
    typedef __attribute__((address_space(3))) int las_int;
    gas_int* gp = (gas_int*)(uintptr_t)gaddr;
    las_int* lp = (las_int*)lds_byte_addr;
    __builtin_amdgcn_global_load_async_to_lds_b32(gp, lp, /*offset=*/0, /*cpol=*/0);
#else
    // GV mode: 64-bit address in VGPR pair, LDS byte address in VGPR.
    asm volatile("global_load_async_to_lds_b32 %0, %1, off"
                 :
                 : "v"(lds_byte_addr), "v"((uint64_t)(uintptr_t)gaddr)
                 : "memory");
#endif
}

__device__ __forceinline__ void wait_asynccnt0() {
#if __has_builtin(__builtin_amdgcn_s_wait_asynccnt)
    __builtin_amdgcn_s_wait_asynccnt(0);
#else
    asm volatile("s_wait_asynccnt 0" ::: "memory");
#endif
}

__global__ __launch_bounds__(NTHREADS)
void Propagation2_64974265254027_kernel(const float* __restrict__ in,
                                        float* __restrict__ out,
                                        int H, int W) {
    __shared__ float lds[LDS_ELEMS];

    const int tid = threadIdx.x;
    const int tx0 = blockIdx.x * TILE_W;
    const int ty0 = blockIdx.y * TILE_H;
    const int b   = blockIdx.z;

    const float* src = in + (size_t)b * H * W;

    // ---- Stage tile + replication halo into LDS via CDNA5 async DMA ----
    // Clamped per-lane addresses materialize the replication pad in LDS once,
    // so the 9-offset phase needs no bounds logic at all. Fully unrolled: all
    // 6 async loads issue back-to-back under ASYNCcnt before the single wait.
#pragma unroll
    for (int i = 0; i < NITER; ++i) {
        const int idx = tid + i * NTHREADS;
        if (idx < LDS_ELEMS) {          // uniform-true for i < 5; EXEC-masked tail
            const int lr = idx / LW;
            const int lc = idx - lr * LW;
            int gy = ty0 + lr - HALO; gy = min(max(gy, 0), H - 1);
            int gx = tx0 + lc - HALO; gx = min(max(gx, 0), W - 1);
            const float* gp = src + (size_t)gy * W + gx;
            const uint32_t laddr = (uint32_t)(uintptr_t)&lds[idx];
            async_f32_to_lds(gp, laddr);
        }
    }
    wait_asynccnt0();     // this wave's async loads landed in LDS
    __syncthreads();      // all waves' loads visible

    // ---- 9 shifted tiles out of LDS, one aligned float4 group per thread ----
    const int row  = tid >> 4;          // 0..15
    const int col4 = (tid & 15) << 2;   // 0,4,...,60
    const int oy   = ty0 + row;
    const int ox   = tx0 + col4;

    // OFFSETS = [(-2,0),(0,-2),(0,0),(0,2),(2,0),(-1,-1),(1,1),(-1,1),(1,-1)]
    constexpr int DY[9] = {-2,  0, 0, 0, 2, -1, 1, -1,  1};
    constexpr int DX[9] = { 0, -2, 0, 2, 0, -1, 1,  1, -1};

    const size_t plane = (size_t)H * W;
    float* outB = out + (size_t)b * 9 * plane + (size_t)oy * W + ox;

#pragma unroll
    for (int k = 0; k < 9; ++k) {
        const float* s = &lds[(row + HALO + DY[k]) * LW + (col4 + HALO + DX[k])];
        v4f v;
        v.x = s[0]; v.y = s[1]; v.z = s[2]; v.w = s[3];
        // Output is written once and never re-read: NT store keeps the 150 MB
        // stream from evicting the L2-resident input.
        __builtin_nontemporal_store(v, (v4f*)(outB + (size_t)k * plane));
    }
}

extern "C" void kernel_launch(void* const* d_in, const int* in_sizes, int n_in,
                              void* d_out, int out_size, void* d_ws, size_t ws_size,
                              hipStream_t stream) {
    const float* in = (const float*)d_in[0];
    float* out = (float*)d_out;

    const int H = 544, W = 960;
    const int B = in_sizes[0] / (H * W);   // 8

    dim3 grid(W / TILE_W, H / TILE_H, B);  // (15, 34, 8) -- exact tiling
    Propagation2_64974265254027_kernel<<<grid, NTHREADS, 0, stream>>>(in, out, H, W);
}